// BiLSTM_54717883351482
// MI455X (gfx1250) — compile-verified
//
#include <hip/hip_runtime.h>
#include <math.h>

// BiLSTM: B=128, T=512, D_IN=512, H=256 (4H=1024), L=2.
// Phase 1 (per layer): Gx = inp @ W_ih^T + b_ih + b_hh  (time-parallel WMMA f32 GEMM)
// Phase 2 (per layer): 512 sequential step launches; h_prev @ W_hh^T via WMMA f32,
//                      + Gx, gates in LDS, length-masked h/c update.
// Phase 3 (per layer): copy final h/c into d_out (order [l0f,l0b,l1f,l1b]).
//
// LDS B-tiles are stored as k-pairs (float2) so every WMMA B-fragment is a single
// aligned ds_load_b64 into an even VGPR pair (no cross-pitch gather / packing movs).

typedef __attribute__((ext_vector_type(2))) float v2f;
typedef __attribute__((ext_vector_type(8))) float v8f;

#define BB   128
#define TT   512
#define DIN  512
#define HH   256
#define G4   1024  // 4*H

#define PP1  144   // phase-1 B pair-pitch (float2 units): 2*144 % 64 == 32
#define PP2  1040  // phase-2 B pair-pitch (float2 units): 2*1040 % 64 == 32

// ---------------------------------------------------------------------------
// Phase 1: Gx GEMM.  WG tile: M=128 rows (r = b*T + t) x N=128 gate cols, K-slices of 32.
// 8 waves; wave w owns M rows [w*16, w*16+16), full N=128 (8 WMMA tiles).
// ---------------------------------------------------------------------------
__global__ __launch_bounds__(256) void gx_gemm_kernel(
    const float* __restrict__ inp,   // (B*T, 512) row-major, r = b*T + t
    const float* __restrict__ Wih,   // layer base: (2, 1024, 512)
    const float* __restrict__ bih,   // layer base: (2, 1024)
    const float* __restrict__ bhh,   // layer base: (2, 1024)
    float* __restrict__ gx)          // (2, T, B, 1024)
{
    __shared__ float  As[128 * 36];   // 128 rows, pitch 36 floats (8B-aligned pairs)
    __shared__ float2 Bp[16 * PP1];   // 16 k-pair rows x 128 cols (pitch 144)

    const int tid  = threadIdx.x;
    const int wave = tid >> 5;       // 0..7 (wave32)
    const int lane = tid & 31;
    const int d    = blockIdx.z;
    const int rowBase = blockIdx.x * 128;
    const int gBase   = blockIdx.y * 128;
    const float* W = Wih + (size_t)d * G4 * DIN;

    const int m  = lane & 15;
    const int kh = (lane >> 4) << 1;          // 0 or 2 (A/B k-pair select)
    const int halfSel = (lane >> 4) << 3;     // 0 or 8 (C row-half select)

    v8f acc[8] = {};

    for (int k0 = 0; k0 < DIN; k0 += 32) {
        __syncthreads();
        // Stage A: 128x32 floats = 1024 float4, 4 per thread
        #pragma unroll
        for (int it = 0; it < 4; ++it) {
            int f4  = tid + it * 256;
            int row = f4 >> 3;
            int c4  = (f4 & 7) << 2;
            const float4 v = *(const float4*)(inp + (size_t)(rowBase + row) * DIN + k0 + c4);
            float* dst = &As[row * 36 + c4];
            dst[0] = v.x; dst[1] = v.y; dst[2] = v.z; dst[3] = v.w;
        }
        // Stage B as k-pairs: Bp[k/2][n] = (W[n][k], W[n][k+1])
        #pragma unroll
        for (int it = 0; it < 4; ++it) {
            int f4 = tid + it * 256;
            int n  = f4 >> 3;
            int kq = (f4 & 7) << 2;
            const float4 v = *(const float4*)(W + (size_t)(gBase + n) * DIN + k0 + kq);
            Bp[((kq >> 1) + 0) * PP1 + n] = make_float2(v.x, v.y);
            Bp[((kq >> 1) + 1) * PP1 + n] = make_float2(v.z, v.w);
        }
        __syncthreads();

        // Prefetch next K-slice while computing this one
        if (k0 + 32 < DIN) {
            {
                int f4  = tid;
                int row = f4 >> 3;
                int c4  = (f4 & 7) << 2;
                __builtin_prefetch(inp + (size_t)(rowBase + row) * DIN + k0 + 32 + c4, 0, 0);
                int n  = f4 >> 3;
                __builtin_prefetch(W + (size_t)(gBase + n) * DIN + k0 + 32, 0, 0);
            }
        }

        #pragma unroll
        for (int kk = 0; kk < 32; kk += 4) {
            v2f a = *(const v2f*)&As[(wave * 16 + m) * 36 + kk + kh];
            #pragma unroll
            for (int nt = 0; nt < 8; ++nt) {
                v2f b = *(const v2f*)&Bp[((kk + kh) >> 1) * PP1 + nt * 16 + m];
                acc[nt] = __builtin_amdgcn_wmma_f32_16x16x4_f32(
                    false, a, false, b, (short)0, acc[nt], false, false);
            }
        }
    }

    // Store with biases, scattering (r = b*T+t) -> gx[d][t][b][g]
    #pragma unroll
    for (int nt = 0; nt < 8; ++nt) {
        int col = gBase + nt * 16 + m;
        float bsum = bih[d * G4 + col] + bhh[d * G4 + col];
        #pragma unroll
        for (int r = 0; r < 8; ++r) {
            int rowL = r + halfSel;
            int gr   = rowBase + wave * 16 + rowL;  // global row = b*T + t
            int bb   = gr >> 9;                     // / 512
            int tt   = gr & 511;
            gx[(((size_t)d * TT + tt) * BB + bb) * G4 + col] = acc[nt][r] + bsum;
        }
    }
}

// ---------------------------------------------------------------------------
// Phase 2: one recurrent step. grid = (B/16, 2 dirs). WG: 16 batch rows x all 1024 gates.
// K=256 over h_prev, in 32 slices of 8.  LDS union (64KB static):
//   during K loop:  hs[16 x 260 floats]  |  B-pairs[4 x 1040 float2]
//   after K loop:   gsm[16 x 1024 floats] (aliases the whole block)
// ---------------------------------------------------------------------------
__global__ __launch_bounds__(256) void lstm_step_kernel(
    const float* __restrict__ gx,     // (2, T, B, 1024) for this layer
    const float* __restrict__ Whh,    // layer base: (2, 1024, 256)
    const int*   __restrict__ lengths,
    float* __restrict__ hbuf,         // (2, B, 256)
    float* __restrict__ cbuf,         // (2, B, 256)
    float* __restrict__ out,          // (B, T, 512); dir d writes cols [d*256, d*256+256)
    int s, int first)
{
    __shared__ float smem[16384];                 // 64KB
    float*  hs  = smem;                           // 16 * 260 floats  (16,640 B)
    float2* Bp  = (float2*)&smem[16 * 260];       // 4 * 1040 float2  (33,280 B)
    float*  gsm = smem;                           // 16 * 1024 floats (reused after K loop)

    const int tid  = threadIdx.x;
    const int wave = tid >> 5;
    const int lane = tid & 31;
    const int d     = blockIdx.y;
    const int bBase = blockIdx.x * 16;
    const int t     = d ? (TT - 1 - s) : s;

    const float* W   = Whh + (size_t)d * G4 * HH;
    const float* gxd = gx + ((size_t)d * TT + t) * BB * G4;
    float* hb = hbuf + (size_t)d * BB * HH;
    float* cb = cbuf + (size_t)d * BB * HH;

    // Stage h_prev tile: 16 x 256 floats = 1024 float4
    #pragma unroll
    for (int it = 0; it < 4; ++it) {
        int f4  = tid + it * 256;
        int row = f4 >> 6;
        int c4  = (f4 & 63) << 2;
        float4 v;
        if (first) v = make_float4(0.f, 0.f, 0.f, 0.f);
        else       v = *(const float4*)(hb + (size_t)(bBase + row) * HH + c4);
        float* dst = &hs[row * 260 + c4];
        dst[0] = v.x; dst[1] = v.y; dst[2] = v.z; dst[3] = v.w;
    }

    const int m  = lane & 15;
    const int kh = (lane >> 4) << 1;
    const int halfSel = (lane >> 4) << 3;

    v8f acc[8] = {};
    for (int k0 = 0; k0 < HH; k0 += 8) {
        __syncthreads();
        // Stage B pairs: Bp[k/2][n] = (W[n][k0+k], W[n][k0+k+1]); 2048 float4, 8/thread
        #pragma unroll
        for (int it = 0; it < 8; ++it) {
            int f4 = tid + it * 256;
            int n  = f4 >> 1;                 // 2 float4 per n-row (8 k)
            int kq = (f4 & 1) << 2;
            const float4 v = *(const float4*)(W + (size_t)n * HH + k0 + kq);
            Bp[((kq >> 1) + 0) * PP2 + n] = make_float2(v.x, v.y);
            Bp[((kq >> 1) + 1) * PP2 + n] = make_float2(v.z, v.w);
        }
        __syncthreads();

        if (k0 + 8 < HH) {
            int f4 = tid;
            int n  = f4 >> 1;
            __builtin_prefetch(W + (size_t)n * HH + k0 + 8, 0, 0);
        }

        #pragma unroll
        for (int kk = 0; kk < 8; kk += 4) {
            v2f a = *(const v2f*)&hs[m * 260 + k0 + kk + kh];
            #pragma unroll
            for (int nt = 0; nt < 8; ++nt) {
                int n = wave * 128 + nt * 16 + m;
                v2f b = *(const v2f*)&Bp[((kk + kh) >> 1) * PP2 + n];
                acc[nt] = __builtin_amdgcn_wmma_f32_16x16x4_f32(
                    false, a, false, b, (short)0, acc[nt], false, false);
            }
        }
    }

    __syncthreads();  // done reading hs/Bp; gsm aliases them

    // g = acc + Gx -> stage gates in LDS
    #pragma unroll
    for (int nt = 0; nt < 8; ++nt) {
        int col = wave * 128 + nt * 16 + m;
        #pragma unroll
        for (int r = 0; r < 8; ++r) {
            int rowL = r + halfSel;
            int bg   = bBase + rowL;
            gsm[rowL * 1024 + col] = acc[nt][r] + gxd[(size_t)bg * G4 + col];
        }
    }
    __syncthreads();

    // Gates + state update: 16 rows x 256 h-cols = 4096 elems, 16 per thread
    #pragma unroll
    for (int j = 0; j < 16; ++j) {
        int flat = tid + j * 256;
        int rowL = flat >> 8;
        int n    = flat & 255;
        int bg   = bBase + rowL;
        float iv = gsm[rowL * 1024 + n];
        float fv = gsm[rowL * 1024 + 256 + n];
        float gv = gsm[rowL * 1024 + 512 + n];
        float ov = gsm[rowL * 1024 + 768 + n];
        iv = 1.f / (1.f + __expf(-iv));
        fv = 1.f / (1.f + __expf(-fv));
        ov = 1.f / (1.f + __expf(-ov));
        gv = tanhf(gv);
        float cp = first ? 0.f : cb[(size_t)bg * HH + n];
        float hp = first ? 0.f : hb[(size_t)bg * HH + n];
        float cn = fv * cp + iv * gv;
        float hn = ov * tanhf(cn);
        bool msk = (t < lengths[bg]);
        hb[(size_t)bg * HH + n] = msk ? hn : hp;
        cb[(size_t)bg * HH + n] = msk ? cn : cp;
        out[((size_t)bg * TT + t) * 512 + d * 256 + n] = msk ? hn : 0.f;
    }
}

// ---------------------------------------------------------------------------
// Phase 3: copy final per-layer state (2, B, 256) into d_out h_n / c_n slots.
// ---------------------------------------------------------------------------
__global__ void copy_state_kernel(const float* __restrict__ hbuf,
                                  const float* __restrict__ cbuf,
                                  float* __restrict__ hout,
                                  float* __restrict__ cout_)
{
    int i = blockIdx.x * 256 + threadIdx.x;   // 0 .. 2*128*256-1
    hout[i]  = hbuf[i];
    cout_[i] = cbuf[i];
}

// ---------------------------------------------------------------------------
extern "C" void kernel_launch(void* const* d_in, const int* in_sizes, int n_in,
                              void* d_out, int out_size, void* d_ws, size_t ws_size,
                              hipStream_t stream)
{
    const float* x       = (const float*)d_in[0];
    const int*   lengths = (const int*)  d_in[1];
    const float* W_ih    = (const float*)d_in[2];
    const float* W_hh    = (const float*)d_in[3];
    const float* b_ih    = (const float*)d_in[4];
    const float* b_hh    = (const float*)d_in[5];
    float* out = (float*)d_out;

    float* ws      = (float*)d_ws;
    float* gx      = ws;                                   // 2*512*128*1024
    float* concat0 = gx + (size_t)2 * TT * BB * G4;        // 128*512*512
    float* hbuf    = concat0 + (size_t)BB * TT * 512;      // 2*128*256
    float* cbuf    = hbuf + (size_t)2 * BB * HH;

    const size_t OUT0 = (size_t)BB * TT * 512;
    float* hN = out + OUT0;
    float* cN = hN + (size_t)4 * BB * HH;

    for (int l = 0; l < 2; ++l) {
        const float* inp   = (l == 0) ? x : concat0;
        const float* Wih_l = W_ih + (size_t)l * 2 * G4 * DIN;
        const float* Whh_l = W_hh + (size_t)l * 2 * G4 * HH;
        const float* bih_l = b_ih + (size_t)l * 2 * G4;
        const float* bhh_l = b_hh + (size_t)l * 2 * G4;
        float* outBuf = (l == 0) ? concat0 : out;

        dim3 g1((BB * TT) / 128, G4 / 128, 2);   // (512, 8, 2)
        gx_gemm_kernel<<<g1, 256, 0, stream>>>(inp, Wih_l, bih_l, bhh_l, gx);

        dim3 g2(BB / 16, 2);                      // (8, 2)
        for (int s = 0; s < TT; ++s) {
            lstm_step_kernel<<<g2, 256, 0, stream>>>(gx, Whh_l, lengths, hbuf, cbuf,
                                                     outBuf, s, (s == 0) ? 1 : 0);
        }
        copy_state_kernel<<<(2 * BB * HH) / 256, 256, 0, stream>>>(
            hbuf, cbuf, hN + (size_t)l * 2 * BB * HH, cN + (size_t)l * 2 * BB * HH);
    }
}